// GraphSAGEPolicy_90898687852768
// MI455X (gfx1250) — compile-verified
//
#include <hip/hip_runtime.h>
#include <hip/hip_bf16.h>

typedef float v2f __attribute__((ext_vector_type(2)));
typedef float v8f __attribute__((ext_vector_type(8)));

#define HID 128

// ---------------- degree / reciprocal ----------------
__global__ __launch_bounds__(256) void deg_kernel(const int* __restrict__ dst,
                                                  float* __restrict__ cnt, int E) {
  int e = blockIdx.x * 256 + threadIdx.x;
  if (e < E) atomicAdd(&cnt[dst[e]], 1.0f);
}

__global__ __launch_bounds__(256) void recip_kernel(const float* __restrict__ cnt,
                                                    float* __restrict__ recip, int N) {
  int i = blockIdx.x * 256 + threadIdx.x;
  if (i < N) recip[i] = 1.0f / fmaxf(cnt[i], 1.0f);
}

// ---------------- edge gather + scatter-add (float4 per thread) ----------------
__global__ __launch_bounds__(256) void scatter_add4(const float* __restrict__ feat,
                                                    const int* __restrict__ src,
                                                    const int* __restrict__ dst,
                                                    float* __restrict__ sums,
                                                    int E, int F, int qshift) {
  unsigned tid = blockIdx.x * 256u + threadIdx.x;
  unsigned total = (unsigned)E << qshift;
  if (tid >= total) return;
  int e = tid >> qshift;
  int q = tid & ((1u << qshift) - 1u);
  int s = src[e];
  int d = dst[e];
  float4 v = *(const float4*)(feat + (size_t)s * F + q * 4);
  float* o = sums + (size_t)d * F + q * 4;
  atomicAdd(o + 0, v.x);
  atomicAdd(o + 1, v.y);
  atomicAdd(o + 2, v.z);
  atomicAdd(o + 3, v.w);
}

// ---------------- scale rows: sums[row][*] *= recip[row] (mean) ----------------
__global__ __launch_bounds__(256) void scale_rows(float* __restrict__ sums,
                                                  const float* __restrict__ recip,
                                                  int N, int F, int qshift) {
  unsigned tid = blockIdx.x * 256u + threadIdx.x;
  unsigned total = (unsigned)N << qshift;
  if (tid >= total) return;
  int row = tid >> qshift;
  int q = tid & ((1u << qshift) - 1u);
  float r = recip[row];
  float4* p = (float4*)(sums + (size_t)row * F + q * 4);
  float4 v = *p;
  v.x *= r; v.y *= r; v.z *= r; v.w *= r;
  *p = v;
}

// ---------------- fused SAGE GEMM: out = [relu]( aggr@Wl^T + b + xin@Wr^T ) --------
// aggr (= sums, pre-scaled by recip) and xin both have row stride K.
// Block = 256 threads = 8 waves = 4 node-groups x 2 col-groups -> 128 nodes x 128 cols.
// Wave: 32 nodes (2 M-tiles) x 64 cols (4 N-tiles); 8 v8f accumulators.
// fp32 WMMA 16x16x4 layouts: A lane m=lane&15, k = kb + 2*(lane>>4) + v;
//                            B lane n=lane&15, same k; D: (r + 8*(lane>>4), lane&15).
// Each loop body loads TWO k-steps of fragments into distinct locals before the
// WMMA chains so loads clause up and overlap with the matrix pipe.
template <int K, int RELU>
__global__ __launch_bounds__(256) void sage_gemm(const float* __restrict__ aggr,
                                                 const float* __restrict__ xin,
                                                 const float* __restrict__ Wl,
                                                 const float* __restrict__ bias,
                                                 const float* __restrict__ Wr,
                                                 float* __restrict__ out,
                                                 int N) {
  const int wave  = threadIdx.x >> 5;
  const int lane  = threadIdx.x & 31;
  const int m     = lane & 15;
  const int khalf = lane >> 4;
  const int ng    = wave >> 1;            // 0..3  node group (32 nodes)
  const int cg    = wave & 1;             // 0..1  col group (64 cols)
  const int node0 = blockIdx.x * 128 + ng * 32;
  const int col0  = cg * 64;

  // A-fragment base pointers for the 2 node tiles (row-clamped; OOB rows only
  // produce D-rows that are never stored, so no masking needed).
  const float* sa[2];
  const float* xa[2];
#pragma unroll
  for (int mt = 0; mt < 2; ++mt) {
    int rr = node0 + mt * 16 + m;
    int row = rr < N ? rr : (N - 1);
    sa[mt] = aggr + (size_t)row * K + khalf * 2;
    xa[mt] = xin  + (size_t)row * K + khalf * 2;
  }
  const float* wl = Wl + (size_t)(col0 + m) * K + khalf * 2;
  const float* wr = Wr + (size_t)(col0 + m) * K + khalf * 2;

  const v8f zero = {0.f, 0.f, 0.f, 0.f, 0.f, 0.f, 0.f, 0.f};
  v8f acc[2][4];
#pragma unroll
  for (int mt = 0; mt < 2; ++mt)
#pragma unroll
    for (int nt = 0; nt < 4; ++nt) acc[mt][nt] = zero;

  for (int kb = 0; kb < K; kb += 8) {
    // ---- load fragments for BOTH k-steps first (independent registers) ----
    v2f a1_0[2], a2_0[2], b1_0[4], b2_0[4];
    v2f a1_1[2], a2_1[2], b1_1[4], b2_1[4];
#pragma unroll
    for (int mt = 0; mt < 2; ++mt) {
      a1_0[mt] = *(const v2f*)(sa[mt] + kb);
      a2_0[mt] = *(const v2f*)(xa[mt] + kb);
      a1_1[mt] = *(const v2f*)(sa[mt] + kb + 4);
      a2_1[mt] = *(const v2f*)(xa[mt] + kb + 4);
    }
#pragma unroll
    for (int nt = 0; nt < 4; ++nt) {
      b1_0[nt] = *(const v2f*)(wl + (size_t)nt * 16 * K + kb);
      b2_0[nt] = *(const v2f*)(wr + (size_t)nt * 16 * K + kb);
      b1_1[nt] = *(const v2f*)(wl + (size_t)nt * 16 * K + kb + 4);
      b2_1[nt] = *(const v2f*)(wr + (size_t)nt * 16 * K + kb + 4);
    }
    // ---- k-step 0: 16 WMMAs ----
#pragma unroll
    for (int mt = 0; mt < 2; ++mt)
#pragma unroll
      for (int nt = 0; nt < 4; ++nt) {
        acc[mt][nt] = __builtin_amdgcn_wmma_f32_16x16x4_f32(
            false, a1_0[mt], false, b1_0[nt], (short)0, acc[mt][nt], false, false);
        acc[mt][nt] = __builtin_amdgcn_wmma_f32_16x16x4_f32(
            false, a2_0[mt], false, b2_0[nt], (short)0, acc[mt][nt], false, false);
      }
    // ---- k-step 1: 16 WMMAs ----
#pragma unroll
    for (int mt = 0; mt < 2; ++mt)
#pragma unroll
      for (int nt = 0; nt < 4; ++nt) {
        acc[mt][nt] = __builtin_amdgcn_wmma_f32_16x16x4_f32(
            false, a1_1[mt], false, b1_1[nt], (short)0, acc[mt][nt], false, false);
        acc[mt][nt] = __builtin_amdgcn_wmma_f32_16x16x4_f32(
            false, a2_1[mt], false, b2_1[nt], (short)0, acc[mt][nt], false, false);
      }
  }

#pragma unroll
  for (int mt = 0; mt < 2; ++mt)
#pragma unroll
    for (int nt = 0; nt < 4; ++nt) {
      const float bv = bias[col0 + nt * 16 + m];
#pragma unroll
      for (int r = 0; r < 8; ++r) {
        float v = acc[mt][nt][r] + bv;
        if (RELU) v = fmaxf(v, 0.0f);
        const int nrow = node0 + mt * 16 + r + 8 * khalf;
        if (nrow < N) out[(size_t)nrow * HID + col0 + nt * 16 + m] = v;
      }
    }
}

// ---------------- global mean pool (partial sums + channel atomics) ----------------
__global__ __launch_bounds__(256) void pool_kernel(const float* __restrict__ h,
                                                   float* __restrict__ g, int N) {
  __shared__ float sh[HID];
  const int c = threadIdx.x & (HID - 1);
  const int half = threadIdx.x >> 7;  // 0 or 1
  float s = 0.0f;
  for (int row = blockIdx.x * 2 + half; row < N; row += gridDim.x * 2)
    s += h[(size_t)row * HID + c];
  if (half == 1) sh[c] = s;
  __syncthreads();
  if (half == 0) atomicAdd(&g[c], s + sh[c]);
}

// ---------------- policy + value heads (one block of 128 threads) ----------------
__global__ __launch_bounds__(128) void heads_kernel(const float* __restrict__ gacc,
                                                    const float* __restrict__ Pw1,
                                                    const float* __restrict__ Pb1,
                                                    const float* __restrict__ Pw2,
                                                    const float* __restrict__ Pb2,
                                                    const float* __restrict__ Vw1,
                                                    const float* __restrict__ Vb1,
                                                    const float* __restrict__ Vw2,
                                                    const float* __restrict__ Vb2,
                                                    float* __restrict__ out, int N) {
  __shared__ float g[HID], p1[HID], v1[HID];
  const int t = threadIdx.x;
  g[t] = gacc[t] / (float)N;
  __syncthreads();
  float sp = Pb1[t];
  float sv = Vb1[t];
  for (int k = 0; k < HID; ++k) {
    sp += Pw1[t * HID + k] * g[k];
    sv += Vw1[t * HID + k] * g[k];
  }
  p1[t] = fmaxf(sp, 0.0f);
  v1[t] = fmaxf(sv, 0.0f);
  __syncthreads();
  if (t < 6) {
    float s = Pb2[t];
    for (int k = 0; k < HID; ++k) s += Pw2[t * HID + k] * p1[k];
    out[t] = s;
  }
  if (t == 6) {
    float s = Vb2[0];
    for (int k = 0; k < HID; ++k) s += Vw2[k] * v1[k];
    out[6] = s;
  }
}

extern "C" void kernel_launch(void* const* d_in, const int* in_sizes, int n_in,
                              void* d_out, int out_size, void* d_ws, size_t ws_size,
                              hipStream_t stream) {
  const int IN = 32;
  const float* x = (const float*)d_in[0];
  const int* ei = (const int*)d_in[1];
  const int N = in_sizes[0] / IN;
  const int E = in_sizes[1] / 2;
  const int* srcp = ei;
  const int* dstp = ei + E;

  const float* W1l = (const float*)d_in[2];
  const float* b1  = (const float*)d_in[3];
  const float* W1r = (const float*)d_in[4];
  const float* W2l = (const float*)d_in[5];
  const float* b2  = (const float*)d_in[6];
  const float* W2r = (const float*)d_in[7];
  const float* W3l = (const float*)d_in[8];
  const float* b3  = (const float*)d_in[9];
  const float* W3r = (const float*)d_in[10];
  const float* Pw1 = (const float*)d_in[11];
  const float* Pb1 = (const float*)d_in[12];
  const float* Pw2 = (const float*)d_in[13];
  const float* Pb2 = (const float*)d_in[14];
  const float* Vw1 = (const float*)d_in[15];
  const float* Vb1 = (const float*)d_in[16];
  const float* Vw2 = (const float*)d_in[17];
  const float* Vb2 = (const float*)d_in[18];

  const size_t NP = ((size_t)N + 127) & ~(size_t)127;
  float* ws    = (float*)d_ws;
  float* cnt   = ws;                    // NP
  float* recip = cnt + NP;              // NP
  float* sums  = recip + NP;            // NP*HID
  float* h1    = sums + NP * HID;       // NP*HID
  float* h2    = h1 + NP * HID;         // NP*HID
  float* gacc  = h2 + NP * HID;         // HID

  const int nodeBlks = (N + 127) / 128;

  // degrees + reciprocals (layer-invariant)
  hipMemsetAsync(cnt, 0, NP * sizeof(float), stream);
  deg_kernel<<<(E + 255) / 256, 256, 0, stream>>>(dstp, cnt, E);
  recip_kernel<<<(N + 255) / 256, 256, 0, stream>>>(cnt, recip, N);

  // ---- layer 1 (F = 32) ----
  hipMemsetAsync(sums, 0, (size_t)N * IN * sizeof(float), stream);
  {
    unsigned total = (unsigned)E << 3;  // 8 float4-quads per edge
    scatter_add4<<<(total + 255) / 256, 256, 0, stream>>>(x, srcp, dstp, sums, E, IN, 3);
    unsigned stot = (unsigned)N << 3;
    scale_rows<<<(stot + 255) / 256, 256, 0, stream>>>(sums, recip, N, IN, 3);
  }
  sage_gemm<32, 1><<<nodeBlks, 256, 0, stream>>>(sums, x, W1l, b1, W1r, h1, N);

  // ---- layer 2 (F = 128) ----
  hipMemsetAsync(sums, 0, (size_t)N * HID * sizeof(float), stream);
  {
    unsigned total = (unsigned)E << 5;  // 32 quads per edge
    scatter_add4<<<(total + 255) / 256, 256, 0, stream>>>(h1, srcp, dstp, sums, E, HID, 5);
    unsigned stot = (unsigned)N << 5;
    scale_rows<<<(stot + 255) / 256, 256, 0, stream>>>(sums, recip, N, HID, 5);
  }
  sage_gemm<128, 1><<<nodeBlks, 256, 0, stream>>>(sums, h1, W2l, b2, W2r, h2, N);

  // ---- layer 3 (F = 128, no relu) ----
  hipMemsetAsync(sums, 0, (size_t)N * HID * sizeof(float), stream);
  {
    unsigned total = (unsigned)E << 5;
    scatter_add4<<<(total + 255) / 256, 256, 0, stream>>>(h2, srcp, dstp, sums, E, HID, 5);
    unsigned stot = (unsigned)N << 5;
    scale_rows<<<(stot + 255) / 256, 256, 0, stream>>>(sums, recip, N, HID, 5);
  }
  sage_gemm<128, 0><<<nodeBlks, 256, 0, stream>>>(sums, h2, W3l, b3, W3r, h1, N);

  // ---- global mean pool + heads ----
  hipMemsetAsync(gacc, 0, HID * sizeof(float), stream);
  pool_kernel<<<240, 256, 0, stream>>>(h1, gacc, N);
  heads_kernel<<<1, 128, 0, stream>>>(gacc, Pw1, Pb1, Pw2, Pb2,
                                      Vw1, Vb1, Vw2, Vb2, (float*)d_out, N);
}